// PhysicsRouter_33148557590991
// MI455X (gfx1250) — compile-verified
//
#include <hip/hip_runtime.h>
#include <math.h>

typedef __attribute__((ext_vector_type(2))) float v2f;
typedef __attribute__((ext_vector_type(8))) float v8f;

#define C_DIM 4096
#define E_DIM 4
#define N_TOK 16384
#define WAVES_PER_BLOCK 4
#define THREADS (WAVES_PER_BLOCK * 32)

__global__ __launch_bounds__(THREADS, 1) void router_main(
    const float* __restrict__ hidden, const float* __restrict__ mass,
    const float* __restrict__ gate_w, const float* __restrict__ mass_bias,
    float* __restrict__ out_logits, float* __restrict__ out_idx,
    float* __restrict__ out_wts, float* __restrict__ g_imp)
{
    // fp32 gate weights, row-major [E][C], plus one zeroed row for dead B lanes
    __shared__ float lds_gate[(E_DIM + 1) * C_DIM];            // 80 KB
    __shared__ float lds_logits[WAVES_PER_BLOCK][16][E_DIM];   // 1 KB
    __shared__ float lds_imp[E_DIM];

    const int tid  = threadIdx.x;
    const int lane = tid & 31;
    const int wave = tid >> 5;

    // Stage gate_w -> LDS (64 KB per block; negligible vs 268 MB hidden stream)
    for (int i = tid; i < E_DIM * C_DIM; i += THREADS)
        lds_gate[i] = gate_w[i];
    for (int i = tid; i < C_DIM; i += THREADS)
        lds_gate[E_DIM * C_DIM + i] = 0.0f;                    // zero row
    if (tid < E_DIM) lds_imp[tid] = 0.0f;
    __syncthreads();

    const int    tile    = blockIdx.x * WAVES_PER_BLOCK + wave; // 16-token tile
    const size_t tokBase = (size_t)tile * 16;

    const int m    = lane & 15;     // A: token row within tile; B: expert column
    const int half = lane >> 4;     // 32-bit WMMA K-pair selector

    // 32-bit A 16x4 layout: lanes 0-15 hold K={0,1}, lanes 16-31 K={2,3}
    const float* aPtr = hidden + (tokBase + (size_t)m) * C_DIM + 2 * half;
    // 32-bit B 4x16 layout mirrors A; dead columns (m >= E) read the zero row
    const float* bPtr = lds_gate + (size_t)((m < E_DIM) ? m : E_DIM) * C_DIM + 2 * half;

    v8f c0 = {};
    v8f c1 = {};
    for (int k0 = 0; k0 < C_DIM; k0 += 32) {
        v2f a[8], b[8];
        #pragma unroll
        for (int s = 0; s < 8; ++s)   // streaming, non-temporal (bypass 192MB L2 residency)
            a[s] = __builtin_nontemporal_load((const v2f*)(aPtr + k0 + 4 * s));
        #pragma unroll
        for (int s = 0; s < 8; ++s)
            b[s] = *(const v2f*)(bPtr + k0 + 4 * s);
        #pragma unroll
        for (int s = 0; s < 8; s += 2) {
            c0 = __builtin_amdgcn_wmma_f32_16x16x4_f32(
                    false, a[s],     false, b[s],     (short)0, c0, false, false);
            c1 = __builtin_amdgcn_wmma_f32_16x16x4_f32(
                    false, a[s + 1], false, b[s + 1], (short)0, c1, false, false);
        }
    }
    v8f c = c0 + c1;

    // C/D layout: vgpr j -> row j + 8*half, col m. Stage valid logits (cols < E).
    if (m < E_DIM) {
        #pragma unroll
        for (int j = 0; j < 8; ++j)
            lds_logits[wave][j + 8 * half][m] = c[j];
    }
    __syncthreads();

    // Per-token epilogue on lanes 0..15 (one token each)
    if (lane < 16) {
        const size_t t = tokBase + lane;
        float lg[E_DIM];
        const float mval = mass[t];
        #pragma unroll
        for (int e = 0; e < E_DIM; ++e)
            lg[e] = lds_logits[wave][lane][e] + mval * mass_bias[e];

        *(float4*)(out_logits + t * 4) = make_float4(lg[0], lg[1], lg[2], lg[3]);

        float mx = fmaxf(fmaxf(lg[0], lg[1]), fmaxf(lg[2], lg[3]));
        float p[E_DIM], s = 0.0f;
        #pragma unroll
        for (int e = 0; e < E_DIM; ++e) { p[e] = __expf(lg[e] - mx); s += p[e]; }
        const float inv = 1.0f / s;
        #pragma unroll
        for (int e = 0; e < E_DIM; ++e) p[e] *= inv;

        // top-2, ties -> lowest index (strict >)
        int i0 = 0; float v0 = p[0];
        #pragma unroll
        for (int e = 1; e < E_DIM; ++e) if (p[e] > v0) { v0 = p[e]; i0 = e; }
        int i1 = -1; float v1 = -1.0f;
        #pragma unroll
        for (int e = 0; e < E_DIM; ++e)
            if (e != i0 && p[e] > v1) { v1 = p[e]; i1 = e; }

        out_idx[t * 2 + 0] = (float)i0;  out_idx[t * 2 + 1] = (float)i1;
        out_wts[t * 2 + 0] = v0;         out_wts[t * 2 + 1] = v1;

        #pragma unroll
        for (int e = 0; e < E_DIM; ++e) atomicAdd(&lds_imp[e], p[e]);
    }
    __syncthreads();
    if (tid < E_DIM) atomicAdd(&g_imp[tid], lds_imp[tid]);
}

__global__ void router_init(float* g_imp) {
    if (threadIdx.x < E_DIM) g_imp[threadIdx.x] = 0.0f;
}

__global__ void router_finalize(const float* __restrict__ g_imp,
                                float* __restrict__ out_aux) {
    if (threadIdx.x == 0) {
        const float target = (float)N_TOK / (float)E_DIM;   // 4096
        float s = 0.0f;
        #pragma unroll
        for (int e = 0; e < E_DIM; ++e) {
            float d = g_imp[e] - target;
            s += d * d;
        }
        *out_aux = s / (float)E_DIM;
    }
}

extern "C" void kernel_launch(void* const* d_in, const int* in_sizes, int n_in,
                              void* d_out, int out_size, void* d_ws, size_t ws_size,
                              hipStream_t stream) {
    const float* hidden    = (const float*)d_in[0];  // [N, C]
    const float* mass      = (const float*)d_in[1];  // [N]
    const float* gate_w    = (const float*)d_in[2];  // [E, C]
    const float* mass_bias = (const float*)d_in[3];  // [E]

    float* out        = (float*)d_out;
    float* out_logits = out;                                     // N*E = 65536
    float* out_idx    = out + (size_t)N_TOK * E_DIM;             // N*2 = 32768
    float* out_aux    = out + (size_t)N_TOK * E_DIM + N_TOK * 2; // 1
    float* out_wts    = out_aux + 1;                             // N*2 = 32768

    float* g_imp = (float*)d_ws;                                 // 4 floats scratch

    (void)in_sizes; (void)n_in; (void)out_size; (void)ws_size;

    router_init<<<1, 64, 0, stream>>>(g_imp);

    const int tiles  = N_TOK / 16;                 // 1024
    const int blocks = tiles / WAVES_PER_BLOCK;    // 256
    router_main<<<blocks, THREADS, 0, stream>>>(hidden, mass, gate_w, mass_bias,
                                                out_logits, out_idx, out_wts, g_imp);

    router_finalize<<<1, 32, 0, stream>>>(g_imp, out_aux);
}